// GConvAdapter_64063732187634
// MI455X (gfx1250) — compile-verified
//
#include <hip/hip_runtime.h>

typedef float v2f __attribute__((ext_vector_type(2)));
typedef float v8f __attribute__((ext_vector_type(8)));

#define HID 256
#define BOT 32

// ---------------- elementwise / degree kernels ----------------

__global__ void k_init_deg(float* __restrict__ deg, int N) {
    int i = blockIdx.x * blockDim.x + threadIdx.x;
    if (i < N) deg[i] = 1.0f;                       // self-loop contributes 1
}

__global__ void k_deg_accum(const long long* __restrict__ col, float* __restrict__ deg, int E) {
    int i = blockIdx.x * blockDim.x + threadIdx.x;
    if (i < E) atomicAdd(&deg[col[i]], 1.0f);
}

__global__ void k_dinv(const float* __restrict__ deg, float* __restrict__ dinv, int N) {
    int i = blockIdx.x * blockDim.x + threadIdx.x;
    if (i < N) {
        float d = deg[i];
        dinv[i] = d > 0.0f ? rsqrtf(d) : 0.0f;
    }
}

__global__ void k_zero(float* __restrict__ p, int n) {
    int i = blockIdx.x * blockDim.x + threadIdx.x;
    if (i < n) p[i] = 0.0f;
}

// h = relu(agg1 + dinv^2 * xd + b_down)   (self-loop folded in analytically)
__global__ void k_hidden(const float* __restrict__ agg1, const float* __restrict__ xd,
                         const float* __restrict__ dinv, const float* __restrict__ bd,
                         float* __restrict__ h, int total) {
    int t = blockIdx.x * blockDim.x + threadIdx.x;
    if (t >= total) return;
    int n = t >> 5;            // node
    int f = t & (BOT - 1);     // feature
    float d = dinv[n];
    float v = agg1[t] + d * d * xd[t] + bd[f];
    h[t] = v > 0.0f ? v : 0.0f;
}

// g = agg2 + dinv^2 * h   (self-loop of second conv)
__global__ void k_selfadd(const float* __restrict__ agg2, const float* __restrict__ h,
                          const float* __restrict__ dinv, float* __restrict__ g, int total) {
    int t = blockIdx.x * blockDim.x + threadIdx.x;
    if (t >= total) return;
    int n = t >> 5;
    float d = dinv[n];
    g[t] = agg2[t] + d * d * h[t];
}

// ---------------- edge scatter (bottleneck dim, 1 wave per edge) ----------------

__global__ void k_scatter(const long long* __restrict__ row, const long long* __restrict__ col,
                          const float* __restrict__ dinv, const float* __restrict__ src,
                          float* __restrict__ dst, int E) {
    int t = blockIdx.x * blockDim.x + threadIdx.x;
    int e = t >> 5;
    if (e >= E) return;
    int f = t & 31;
    long long r = row[e];
    long long c = col[e];
    float w = dinv[r] * dinv[c];                    // wave-uniform
    atomicAdd(dst + (size_t)c * BOT + f, src[(size_t)r * BOT + f] * w);
}

// ---------------- WMMA GEMMs (V_WMMA_F32_16X16X4_F32) ----------------
// A 16x4 layout: lanes 0-15 hold M=lane; VGPR0/1 = K pair (k, k+1), K base = 2*(lane>>4)
// B 4x16 layout (mirror): lanes hold N=lane&15; VGPR0/1 = K pair, K base = 2*(lane>>4)
// C/D 16x16:  VGPR r holds M = r + 8*(lane>>4), N = lane&15

// xd[N][32] = x[N][256] @ W_down[32][256]^T
__global__ void k_gemm_down(const float* __restrict__ x, const float* __restrict__ Wd,
                            float* __restrict__ xd, int N) {
    const int lane  = threadIdx.x & 31;
    const int wave  = threadIdx.x >> 5;
    const int m0    = (blockIdx.x * 4 + wave) * 16;
    if (m0 >= N) return;                            // wave-uniform, EXEC stays all-1s
    const int n0    = blockIdx.y * 16;
    const int khalf = (lane >> 4) * 2;
    const float* xrow = x  + (size_t)(m0 + (lane & 15)) * HID;
    const float* wrow = Wd + (size_t)(n0 + (lane & 15)) * HID;   // B[k][n] = Wd[n][k]
    v8f acc = {};
    for (int k0 = 0; k0 < HID; k0 += 4) {
        v2f a, b;
        a.x = xrow[k0 + khalf];
        a.y = xrow[k0 + khalf + 1];
        b.x = wrow[k0 + khalf];
        b.y = wrow[k0 + khalf + 1];
        acc = __builtin_amdgcn_wmma_f32_16x16x4_f32(false, a, false, b,
                                                    (short)0, acc, false, false);
    }
    const int nst   = n0 + (lane & 15);
    const int mbase = m0 + (lane >> 4) * 8;
#pragma unroll
    for (int r = 0; r < 8; ++r)
        xd[(size_t)(mbase + r) * BOT + nst] = acc[r];
}

// out[N][256] = g[N][32] @ W_up[256][32]^T + b_up + x   (bias + residual fused)
__global__ void k_gemm_up(const float* __restrict__ g, const float* __restrict__ Wu,
                          const float* __restrict__ bu, const float* __restrict__ x,
                          float* __restrict__ out, int N) {
    const int lane  = threadIdx.x & 31;
    const int wave  = threadIdx.x >> 5;
    const int m0    = (blockIdx.x * 4 + wave) * 16;
    if (m0 >= N) return;
    const int n0    = blockIdx.y * 16;
    const int khalf = (lane >> 4) * 2;
    const float* grow = g  + (size_t)(m0 + (lane & 15)) * BOT;
    const float* wrow = Wu + (size_t)(n0 + (lane & 15)) * BOT;   // B[k][n] = Wu[n][k]
    v8f acc = {};
#pragma unroll
    for (int k0 = 0; k0 < BOT; k0 += 4) {
        v2f a, b;
        a.x = grow[k0 + khalf];
        a.y = grow[k0 + khalf + 1];
        b.x = wrow[k0 + khalf];
        b.y = wrow[k0 + khalf + 1];
        acc = __builtin_amdgcn_wmma_f32_16x16x4_f32(false, a, false, b,
                                                    (short)0, acc, false, false);
    }
    const int nst   = n0 + (lane & 15);
    const float bias = bu[nst];
    const int mbase = m0 + (lane >> 4) * 8;
#pragma unroll
    for (int r = 0; r < 8; ++r) {
        size_t o = (size_t)(mbase + r) * HID + nst;
        out[o] = acc[r] + bias + x[o];
    }
}

// ---------------- launch ----------------

extern "C" void kernel_launch(void* const* d_in, const int* in_sizes, int n_in,
                              void* d_out, int out_size, void* d_ws, size_t ws_size,
                              hipStream_t stream) {
    (void)n_in; (void)out_size; (void)ws_size;

    const float*     x   = (const float*)d_in[0];
    const long long* ei  = (const long long*)d_in[1];   // [2][E], int64
    const float*     Wd  = (const float*)d_in[2];       // [32][256]
    const float*     bd  = (const float*)d_in[3];       // [32]
    const float*     Wu  = (const float*)d_in[4];       // [256][32]
    const float*     bu  = (const float*)d_in[5];       // [256]
    float*           out = (float*)d_out;

    const int N = in_sizes[0] / HID;                    // 100000
    const int E = in_sizes[1] / 2;                      // 800000
    const long long* row = ei;
    const long long* col = ei + E;

    char* ws = (char*)d_ws;
    float* deg  = (float*)ws;  ws += (size_t)N * sizeof(float);
    float* dinv = (float*)ws;  ws += (size_t)N * sizeof(float);
    float* xd   = (float*)ws;  ws += (size_t)N * BOT * sizeof(float);
    float* agg1 = (float*)ws;  ws += (size_t)N * BOT * sizeof(float);
    float* h    = (float*)ws;  ws += (size_t)N * BOT * sizeof(float);
    float* agg2 = (float*)ws;  ws += (size_t)N * BOT * sizeof(float);
    float* gbuf = (float*)ws;  ws += (size_t)N * BOT * sizeof(float);

    const int NB   = N * BOT;
    const int TPB  = 256;
    const dim3 gDown((unsigned)((N + 63) / 64), BOT / 16);
    const dim3 gUp  ((unsigned)((N + 63) / 64), HID / 16);
    const unsigned edgeBlocks = (unsigned)(((long long)E * 32 + TPB - 1) / TPB);

    // normalization
    k_init_deg <<<(N + TPB - 1) / TPB, TPB, 0, stream>>>(deg, N);
    k_deg_accum<<<(E + TPB - 1) / TPB, TPB, 0, stream>>>(col, deg, E);
    k_dinv     <<<(N + TPB - 1) / TPB, TPB, 0, stream>>>(deg, dinv, N);

    // conv1: project down (WMMA), aggregate in 32-dim, relu+bias+self-loop
    k_gemm_down<<<gDown, 128, 0, stream>>>(x, Wd, xd, N);
    k_zero     <<<(NB + TPB - 1) / TPB, TPB, 0, stream>>>(agg1, NB);
    k_scatter  <<<edgeBlocks, TPB, 0, stream>>>(row, col, dinv, xd, agg1, E);
    k_hidden   <<<(NB + TPB - 1) / TPB, TPB, 0, stream>>>(agg1, xd, dinv, bd, h, NB);

    // conv2: aggregate FIRST in 32-dim (associativity), then project up (WMMA) + bias + skip
    k_zero     <<<(NB + TPB - 1) / TPB, TPB, 0, stream>>>(agg2, NB);
    k_scatter  <<<edgeBlocks, TPB, 0, stream>>>(row, col, dinv, h, agg2, E);
    k_selfadd  <<<(NB + TPB - 1) / TPB, TPB, 0, stream>>>(agg2, h, dinv, gbuf, NB);
    k_gemm_up  <<<gUp, 128, 0, stream>>>(gbuf, Wu, bu, x, out, N);
}